// TGCN_38620345925848
// MI455X (gfx1250) — compile-verified
//
#include <hip/hip_runtime.h>

// ---------------------------------------------------------------------------
// Problem constants (from reference setup_inputs)
// ---------------------------------------------------------------------------
#define BS      8
#define NODE    1024
#define IN_DIM  64
#define OUT_DIM 64
#define D_E     10
#define T_D     6
#define EMBED   16
#define CHEB_K  3
#define KI      (CHEB_K * IN_DIM)   // 192

typedef float v2f __attribute__((ext_vector_type(2)));
typedef float v8f __attribute__((ext_vector_type(8)));

// ---------------------------------------------------------------------------
// CDNA5 async global->LDS staging (ASYNCcnt path), with safe fallback.
// Probe-verified signature (from hipcc diagnostic):
//   __builtin_amdgcn_global_load_async_to_lds_b128(
//       v4i addrspace(1)* gsrc, v4i addrspace(3)* ldst, imm int off, imm int cpol)
// ---------------------------------------------------------------------------
#if __has_builtin(__builtin_amdgcn_global_load_async_to_lds_b128)
#define USE_ASYNC_LDS 1
#endif

typedef int b128_t __attribute__((vector_size(4 * sizeof(int))));
typedef __attribute__((address_space(1))) b128_t* gas_b128_ptr;
typedef __attribute__((address_space(3))) b128_t* las_b128_ptr;

__device__ __forceinline__ void stage_b128(const float* __restrict__ g, float* l) {
#ifdef USE_ASYNC_LDS
    // generic->AS1 is numerically identity; generic shared->AS3 keeps low 32
    // bits (LDS aperture places the byte offset in addr[31:0]).
    __builtin_amdgcn_global_load_async_to_lds_b128(
        (gas_b128_ptr)(unsigned long long)g,
        (las_b128_ptr)(unsigned)(unsigned long long)l,
        /*offset=*/0, /*cpol=*/0);
#else
    const float2* gs = (const float2*)g;     // 8B-aligned fallback (2x b64)
    float2* ls = (float2*)l;
    float2 v0 = gs[0];
    float2 v1 = gs[1];
    ls[0] = v0;
    ls[1] = v1;
#endif
}

__device__ __forceinline__ void stage_wait() {
#ifdef USE_ASYNC_LDS
#if __has_builtin(__builtin_amdgcn_s_wait_asynccnt)
    __builtin_amdgcn_s_wait_asynccnt(0);
#else
    asm volatile("s_wait_asynccnt 0" ::: "memory");
#endif
#endif
}

// ---------------------------------------------------------------------------
// K1: a[n,m] = <node_emb[n,:], node_emb[m,:]>   (1024x1024, d_e=10)
// ---------------------------------------------------------------------------
__global__ void gram_kernel(const float* __restrict__ ne, float* __restrict__ a) {
    int idx = blockIdx.x * blockDim.x + threadIdx.x;   // NODE*NODE threads
    int n = idx >> 10;
    int m = idx & 1023;
    float s = 0.f;
#pragma unroll
    for (int d = 0; d < D_E; ++d)
        s += ne[n * D_E + d] * ne[m * D_E + d];
    a[idx] = s;
}

// ---------------------------------------------------------------------------
// K2: A[b,n,:] = softmax(relu(a[n,:] + <n_t[b],t[b]>))   one block per (b,n)
// ---------------------------------------------------------------------------
__global__ __launch_bounds__(256)
void softmax_kernel(const float* __restrict__ a, const float* __restrict__ t,
                    const float* __restrict__ nt, float* __restrict__ A) {
    int b = blockIdx.x >> 10;
    int n = blockIdx.x & 1023;

    float at = 0.f;
#pragma unroll
    for (int d = 0; d < T_D; ++d)
        at += nt[b * T_D + d] * t[b * T_D + d];

    const float* row = a + (size_t)n * NODE;
    int tid  = threadIdx.x;                 // 256 threads, 4 elems each
    int wave = tid >> 5;
    int lane = tid & 31;

    float v[4];
    float mx = 0.f;                          // relu => max >= 0
#pragma unroll
    for (int j = 0; j < 4; ++j) {
        float x = row[tid + 256 * j] + at;
        x = x > 0.f ? x : 0.f;
        v[j] = x;
        mx = fmaxf(mx, x);
    }
    __shared__ float sm[8];
    __shared__ float ss[8];
    for (int off = 16; off > 0; off >>= 1)
        mx = fmaxf(mx, __shfl_xor(mx, off, 32));
    if (lane == 0) sm[wave] = mx;
    __syncthreads();
    mx = sm[0];
#pragma unroll
    for (int w = 1; w < 8; ++w) mx = fmaxf(mx, sm[w]);

    float sum = 0.f;
#pragma unroll
    for (int j = 0; j < 4; ++j) { v[j] = __expf(v[j] - mx); sum += v[j]; }
    for (int off = 16; off > 0; off >>= 1)
        sum += __shfl_xor(sum, off, 32);
    if (lane == 0) ss[wave] = sum;
    __syncthreads();
    sum = 0.f;
#pragma unroll
    for (int w = 0; w < 8; ++w) sum += ss[w];
    float inv = 1.f / sum;

    float* out = A + ((size_t)b * NODE + n) * NODE;
#pragma unroll
    for (int j = 0; j < 4; ++j) out[tid + 256 * j] = v[j] * inv;
}

// ---------------------------------------------------------------------------
// K3: repack weights_pool[d,k,i,o] -> WpR[k*64+i, d*64+o]  (192 x 1024)
// ---------------------------------------------------------------------------
__global__ void pack_w_kernel(const float* __restrict__ wp, float* __restrict__ wr) {
    int idx = blockIdx.x * blockDim.x + threadIdx.x;   // 192*1024
    int r = idx >> 10, c = idx & 1023;
    int k = r >> 6, i = r & 63, d = c >> 6, o = c & 63;
    wr[idx] = wp[((d * CHEB_K + k) << 12) + (i << 6) + o];
}

// K4: Xg[:, 0:64] = x   (identity Chebyshev term)
__global__ void copy_x_kernel(const float* __restrict__ x, float* __restrict__ xg) {
    int idx = blockIdx.x * blockDim.x + threadIdx.x;   // BS*NODE*IN_DIM
    int bn = idx >> 6, i = idx & 63;
    xg[(size_t)bn * KI + i] = x[idx];
}

// ---------------------------------------------------------------------------
// Core batched GEMM using V_WMMA_F32_16X16X4_F32:
//   D = alpha * (A @ B) + beta * C      (row-major, per-z pointer strides)
//
// Block = 256 threads (8 waves, wave32). Block tile 32x64; each wave owns one
// 16x16 f32 accumulator (8 VGPRs). A staged through double-buffered LDS in
// 32x32 chunks via async global->LDS b128 (coalesced), fragments read back as
// conflict-free ds_load_b64 (row stride 36 words: banks 0..63 all distinct,
// rows 16B aligned). B read directly with compile-time immediate offsets.
//
// ISA fragment layouts (CDNA5 ISA):
//   A 16x4 : lanes 0-15 -> K{0,1}, lanes 16-31 -> K{2,3} (one float2/lane)
//   B 4x16 : VGPR0 = rows K0|K2 striped over lanes, VGPR1 = rows K1|K3
//   C/D    : VGPR r -> M=r (lanes 0-15), M=r+8 (lanes 16-31), N=lane%16
// ---------------------------------------------------------------------------
template<int LDA, int LDB, int LDC, int LDD, int K, bool HAS_C>
__global__ __launch_bounds__(256)
void wmma_gemm_f32(const float* __restrict__ Ab, long long sA,
                   const float* __restrict__ Bb, long long sB,
                   const float* __restrict__ Cb, long long sC,
                   float* __restrict__ Db, long long sD,
                   float alpha, float beta)
{
    constexpr int KCHUNK = 32;
    __shared__ float As[2][32][36];          // padded: conflict-free + 16B rows

    int z = blockIdx.z;
    const float* Am = Ab + (size_t)z * sA;
    const float* Bm = Bb + (size_t)z * sB;
    const float* Cm = HAS_C ? (Cb + (size_t)z * sC) : nullptr;
    float*       Dm = Db + (size_t)z * sD;

    int tid  = threadIdx.x;
    int wave = tid >> 5;
    int lane = tid & 31;
    int half = lane >> 4;                    // 0: K pair {0,1}, 1: {2,3}
    int l    = lane & 15;

    int mblk = blockIdx.y * 32;
    int m0   = mblk + (wave >> 2) * 16;      // row-tile origin
    int n0   = blockIdx.x * 64 + (wave & 3) * 16;  // col-tile origin
    int arow = (wave >> 2) * 16 + l;         // this lane's A row in LDS

    // Cooperative staging map: thread -> 4 consecutive floats of 32x32 chunk
    int srow = tid >> 3;                     // 0..31
    int scol = (tid & 7) * 4;                // 0,4,...,28
    const float* ag = Am + (size_t)(mblk + srow) * LDA + scol;
    float* as0 = &As[0][srow][scol];
    float* as1 = &As[1][srow][scol];

    // B base: immediate offsets kk*LDB inside chunk, bump per chunk
    const float* bpc = Bm + (size_t)(2 * half) * LDB + n0 + l;

    v8f acc = {};

    // Prologue: stage chunk 0 into buffer 0
    stage_b128(ag, as0);
    stage_wait();
    __syncthreads();

    for (int kc = 0; kc < K; kc += KCHUNK) {
        int buf = (kc >> 5) & 1;
        if (kc + KCHUNK < K)                 // prefetch next chunk into buf^1
            stage_b128(ag + kc + KCHUNK, buf ? as0 : as1);

#pragma unroll
        for (int kk = 0; kk < KCHUNK; kk += 4) {
            v2f afrag = *(const v2f*)&As[buf][arow][kk + 2 * half];
            v2f bfrag;
            bfrag.x = bpc[(size_t)kk * LDB];
            bfrag.y = bpc[(size_t)kk * LDB + LDB];
            acc = __builtin_amdgcn_wmma_f32_16x16x4_f32(
                      /*neg_a=*/false, afrag, /*neg_b=*/false, bfrag,
                      /*c_mod=*/(short)0, acc, /*reuse_a=*/false, /*reuse_b=*/false);
        }
        bpc += (size_t)KCHUNK * LDB;
        stage_wait();
        __syncthreads();
    }

#pragma unroll
    for (int r = 0; r < 8; ++r) {
        int row = m0 + r + 8 * half;
        int col = n0 + l;
        float v = alpha * acc[r];
        if constexpr (HAS_C) v += beta * Cm[(size_t)row * LDC + col];
        Dm[(size_t)row * LDD + col] = v;
    }
}

// ---------------------------------------------------------------------------
// K5: out[b,n,o] = sum_d emb[b,n,d] * (y[b,n,d*64+o] + bias_pool[d,o])
//     emb = concat(node_emb[n,0:10], n_t[b,0:6])
// ---------------------------------------------------------------------------
__global__ void final_kernel(const float* __restrict__ y,
                             const float* __restrict__ ne,
                             const float* __restrict__ nt,
                             const float* __restrict__ bias_pool,
                             float* __restrict__ out) {
    int idx = blockIdx.x * blockDim.x + threadIdx.x;   // BS*NODE*OUT_DIM
    int o = idx & 63;
    int n = (idx >> 6) & 1023;
    int b = idx >> 16;
    const float* yrow = y + ((size_t)(b * NODE + n) << 10);   // 1024 per (b,n)
    float s = 0.f;
#pragma unroll
    for (int d = 0; d < EMBED; ++d) {
        float e = (d < D_E) ? ne[n * D_E + d] : nt[b * T_D + (d - D_E)];
        s += e * (yrow[(d << 6) + o] + bias_pool[(d << 6) + o]);
    }
    out[idx] = s;
}

// ---------------------------------------------------------------------------
// Launcher
// ---------------------------------------------------------------------------
extern "C" void kernel_launch(void* const* d_in, const int* in_sizes, int n_in,
                              void* d_out, int out_size, void* d_ws, size_t ws_size,
                              hipStream_t stream) {
    const float* x   = (const float*)d_in[0];  // [8,1024,64]
    const float* ne  = (const float*)d_in[1];  // [1024,10]
    const float* t   = (const float*)d_in[2];  // [8,6]
    const float* nt  = (const float*)d_in[3];  // [8,6]
    const float* wp  = (const float*)d_in[5];  // [16,3,64,64]
    const float* bp  = (const float*)d_in[6];  // [16,64]
    float* out = (float*)d_out;                // [8,1024,64]

    // Workspace layout (floats), all 16B aligned
    float* ws   = (float*)d_ws;
    float* a    = ws;                               // 1024*1024
    float* A    = a   + (size_t)NODE * NODE;        // 8*1024*1024
    float* Xg   = A   + (size_t)BS * NODE * NODE;   // 8*1024*192
    float* WpR  = Xg  + (size_t)BS * NODE * KI;     // 192*1024
    float* y    = WpR + (size_t)KI * NODE;          // 8*1024*1024
    // total ~78.4 MB

    // 1) Gram matrix (shared across batch)
    gram_kernel<<<(NODE * NODE) / 256, 256, 0, stream>>>(ne, a);

    // 2) relu + softmax per (b, row)
    softmax_kernel<<<BS * NODE, 256, 0, stream>>>(a, t, nt, A);

    // 3) weight repack + identity term copy
    pack_w_kernel<<<(KI * NODE) / 256, 256, 0, stream>>>(wp, WpR);
    copy_x_kernel<<<(BS * NODE * IN_DIM) / 256, 256, 0, stream>>>(x, Xg);

    // 4) xg1 = A @ x      -> Xg[:, 64:128]
    {
        dim3 grid(IN_DIM / 64, NODE / 32, BS);
        wmma_gemm_f32<NODE, IN_DIM, 1, KI, NODE, false><<<grid, 256, 0, stream>>>(
            A,        (long long)NODE * NODE,
            x,        (long long)NODE * IN_DIM,
            nullptr,  0,
            Xg + 64,  (long long)NODE * KI,
            1.0f, 0.0f);
    }

    // 5) xg2 = 2*A @ xg1 - x   -> Xg[:, 128:192]
    {
        dim3 grid(IN_DIM / 64, NODE / 32, BS);
        wmma_gemm_f32<NODE, KI, IN_DIM, KI, NODE, true><<<grid, 256, 0, stream>>>(
            A,        (long long)NODE * NODE,
            Xg + 64,  (long long)NODE * KI,
            x,        (long long)NODE * IN_DIM,
            Xg + 128, (long long)NODE * KI,
            2.0f, -1.0f);
    }

    // 6) y = Xg @ WpR     [8192 x 192] @ [192 x 1024]
    {
        dim3 grid(NODE / 64, (BS * NODE) / 32, 1);
        wmma_gemm_f32<KI, NODE, 1, NODE, KI, false><<<grid, 256, 0, stream>>>(
            Xg,   0,
            WpR,  0,
            nullptr, 0,
            y,    0,
            1.0f, 0.0f);
    }

    // 7) epilogue: embedding-weighted reduction + bias
    final_kernel<<<(BS * NODE * OUT_DIM) / 256, 256, 0, stream>>>(y, ne, nt, bp, out);
}